// PDMatchingLoss_17566416241243
// MI455X (gfx1250) — compile-verified
//
#include <hip/hip_runtime.h>
#include <stdint.h>

typedef unsigned int u32;
typedef unsigned long long u64;

#define SDIM 1026
#define IMGS 32            // 16 input + 16 target images
#define PAIRS (IMGS*3)     // 3 selections per image
#define KSEL 256
#define ROWS 6
#define TILES ((SDIM + ROWS - 1) / ROWS)   // 171
#define IMG_ELEMS (1024*1024)

// ---- workspace layout (bytes). total < 0.5 MB ----
#define O_HIST   0          // PAIRS*256*4 = 98304
#define O_PREFIX 98304      // PAIRS*8     = 768
#define O_KREM   99072      // PAIRS*4
#define O_CNT    99456      // PAIRS*4
#define O_GMAX   99840      // IMGS*4
#define O_KEYS   100096     // PAIRS*256*8 = 196608   (8-aligned)
#define O_VALS   296704     // PAIRS*256*4
#define O_PIDX   395008     // PAIRS*256*4

typedef __attribute__((ext_vector_type(4))) u32 v4u_t;
typedef __attribute__((ext_vector_type(8))) int v8i_t;
typedef __attribute__((ext_vector_type(4))) int v4i_t;

__device__ __forceinline__ int imin(int a, int b){ return a < b ? a : b; }
__device__ __forceinline__ int imax(int a, int b){ return a > b ? a : b; }

// order-preserving float32 -> uint32 (total order)
__device__ __forceinline__ u32 monokey(float f){
  u32 u = __float_as_uint(f);
  return (u & 0x80000000u) ? ~u : (u | 0x80000000u);
}

// ---- TDM staging: raw input rows [rowbase .. rowbase+nrows) x 1024 -> LDS tile ----
__device__ __forceinline__ void stage_tile_tdm(float* tile, const float* rowbase, int nrows){
  if ((threadIdx.x >> 5) == 0) {   // one wave issues; TDM ignores EXEC
    u64 gaddr = (u64)(uintptr_t)rowbase;
    u32 lds   = (u32)(uintptr_t)tile;   // LDS aperture: addr[31:0] = wave-relative LDS byte offset
    v4u_t g0; v8i_t g1; v4i_t gz4; v8i_t gz8;
    gz4[0]=0; gz4[1]=0; gz4[2]=0; gz4[3]=0;
    gz8[0]=0; gz8[1]=0; gz8[2]=0; gz8[3]=0; gz8[4]=0; gz8[5]=0; gz8[6]=0; gz8[7]=0;
    // D# group0: count=1 | lds_addr | global_addr[56:0] | type=2
    g0[0] = 1u;
    g0[1] = lds;
    g0[2] = (u32)(gaddr & 0xFFFFFFFFull);
    g0[3] = (u32)((gaddr >> 32) & 0x1FFFFFFull) | (2u << 30);
    // D# group1: data_size=4B; tensor_dim0=1024 @[79:48]; tensor_dim1=nrows @[111:80];
    //            tile_dim0=1024 @[127:112]; tile_dim1=nrows @[143:128]; stride0=1024 @[207:160]
    g1[0] = (int)(2u << 16);
    g1[1] = (int)(1024u << 16);
    g1[2] = (int)((u32)nrows << 16);
    g1[3] = (int)(1024u << 16);
    g1[4] = nrows;
    g1[5] = 1024;
    g1[6] = 0;
    g1[7] = 0;
    __builtin_amdgcn_tensor_load_to_lds(g0, g1, gz4, gz4, gz8, 0);
    __builtin_amdgcn_s_wait_tensorcnt(0);
  }
  __syncthreads();
}

// prepped value v(y,x): border -> 0, interior -> 1 - clamp(raw)
__device__ __forceinline__ float getv(const float* tile, int rawLo, int y, int x){
  if (y == 0 || y == SDIM-1 || x == 0 || x == SDIM-1) return 0.0f;
  float r = tile[(y - 1 - rawLo)*1024 + (x - 1)];
  r = fminf(fmaxf(r, 0.0f), 1.0f);
  return 1.0f - r;
}

// composite selection key; 0 == not a candidate
__device__ __forceinline__ u64 pixel_key(const float* tile, int rawLo, int y, int x, int sel){
  float vc = getv(tile, rawLo, y, x);
  float nmin = 3.0e38f, nmax = -3.0e38f;
  if (y > 0)      { float t = getv(tile,rawLo,y-1,x); nmin=fminf(nmin,t); nmax=fmaxf(nmax,t); }
  if (y < SDIM-1) { float t = getv(tile,rawLo,y+1,x); nmin=fminf(nmin,t); nmax=fmaxf(nmax,t); }
  if (x > 0)      { float t = getv(tile,rawLo,y,x-1); nmin=fminf(nmin,t); nmax=fmaxf(nmax,t); }
  if (x < SDIM-1) { float t = getv(tile,rawLo,y,x+1); nmin=fminf(nmin,t); nmax=fmaxf(nmax,t); }
  bool cand = (sel == 0) ? (vc <= nmin) : (vc >= nmax);
  if (!cand) return 0ull;
  u32 m  = monokey(vc);
  u32 vk = (sel == 1) ? m : ~m;            // sel1: value descending; sel0/2: value ascending
  u32 flat = (u32)(y*SDIM + x);
  return ((u64)vk << 21) | ((~flat) & 0x1FFFFFu);   // ties: smaller index = larger key
}

__global__ __launch_bounds__(256) void init_kernel(u32* hist, u64* prefix, u32* krem, u32* cnt, u32* gmax){
  int i = blockIdx.x*256 + threadIdx.x;
  if (i < PAIRS*256) hist[i] = 0;
  if (i < PAIRS){ prefix[i] = 0ull; krem[i] = KSEL; cnt[i] = 0; }
  if (i < IMGS) gmax[i] = 0x80000000u;     // monokey(+0.0f): border value
}

__global__ __launch_bounds__(256) void gmax_kernel(const float* __restrict__ in, const float* __restrict__ tg, u32* gmax){
  __shared__ u32 red[256];
  int img = blockIdx.y;
  const float* base = (img < 16) ? in + (size_t)img*IMG_ELEMS : tg + (size_t)(img-16)*IMG_ELEMS;
  int i = blockIdx.x*256 + threadIdx.x;
  float r = base[i];
  r = fminf(fmaxf(r, 0.0f), 1.0f);
  red[threadIdx.x] = __float_as_uint(1.0f - r) | 0x80000000u;
  __syncthreads();
  for (int s2 = 128; s2 > 0; s2 >>= 1){
    if (threadIdx.x < s2) red[threadIdx.x] = red[threadIdx.x] > red[threadIdx.x+s2] ? red[threadIdx.x] : red[threadIdx.x+s2];
    __syncthreads();
  }
  if (threadIdx.x == 0) atomicMax(&gmax[img], red[0]);
}

__global__ __launch_bounds__(256) void hist_kernel(const float* __restrict__ in, const float* __restrict__ tg,
                                                   u32* __restrict__ hist, const u64* __restrict__ prefix, int shift){
  __shared__ float tile[8*1024];
  __shared__ u32 lhist[256];
  int p = blockIdx.y;
  int img = p / 3, sel = p - img*3;
  const float* base = (img < 16) ? in + (size_t)img*IMG_ELEMS : tg + (size_t)(img-16)*IMG_ELEMS;
  int r0 = blockIdx.x * ROWS;
  int rawLo = imax(r0 - 2, 0), rawHi = imin(r0 + ROWS - 1, 1023);
  int nrows = rawHi - rawLo + 1;
  lhist[threadIdx.x] = 0;
  stage_tile_tdm(tile, base + (size_t)rawLo*1024, nrows);
  u64 pref = prefix[p];
  int hi = shift + 8;
  int yEnd = imin(r0 + ROWS, SDIM);
  for (int y = r0; y < yEnd; ++y)
    for (int x = threadIdx.x; x < SDIM; x += 256){
      u64 key = pixel_key(tile, rawLo, y, x, sel);
      if (key != 0ull && ((key >> hi) == (pref >> hi)))
        atomicAdd(&lhist[(u32)(key >> shift) & 255u], 1u);
    }
  __syncthreads();
  u32 c = lhist[threadIdx.x];
  if (c) atomicAdd(&hist[p*256 + threadIdx.x], c);
}

__global__ __launch_bounds__(256) void scan_kernel(u32* hist, u64* prefix, u32* krem, int shift){
  __shared__ u32 h[256];
  int p = blockIdx.x, t = threadIdx.x;
  h[t] = hist[p*256 + t];
  hist[p*256 + t] = 0;               // reset for next pass
  __syncthreads();
  if (t == 0){
    u32 kr = krem[p];
    u64 cum = 0; int d = 0; u32 nk = kr;
    for (int b = 255; b >= 0; --b){
      cum += h[b];
      if (cum >= (u64)kr){ d = b; nk = kr - (u32)(cum - h[b]); break; }
    }
    prefix[p] |= ((u64)d) << shift;
    krem[p] = nk;
  }
}

__global__ __launch_bounds__(256) void collect_kernel(const float* __restrict__ in, const float* __restrict__ tg,
                                                      const u64* __restrict__ prefix, u32* cnt, u64* keys){
  __shared__ float tile[8*1024];
  int p = blockIdx.y;
  int img = p / 3, sel = p - img*3;
  const float* base = (img < 16) ? in + (size_t)img*IMG_ELEMS : tg + (size_t)(img-16)*IMG_ELEMS;
  int r0 = blockIdx.x * ROWS;
  int rawLo = imax(r0 - 2, 0), rawHi = imin(r0 + ROWS - 1, 1023);
  int nrows = rawHi - rawLo + 1;
  stage_tile_tdm(tile, base + (size_t)rawLo*1024, nrows);
  u64 thresh = prefix[p];            // exact K-th largest key (keys are unique)
  int yEnd = imin(r0 + ROWS, SDIM);
  for (int y = r0; y < yEnd; ++y)
    for (int x = threadIdx.x; x < SDIM; x += 256){
      u64 key = pixel_key(tile, rawLo, y, x, sel);
      if (key != 0ull && key >= thresh){
        u32 pos = atomicAdd(&cnt[p], 1u);
        if (pos < KSEL) keys[p*KSEL + pos] = key;
      }
    }
}

__global__ __launch_bounds__(256) void sort_kernel(const u64* __restrict__ keys, const u32* __restrict__ cnt,
                                                   float* __restrict__ vals, u32* __restrict__ pidx){
  __shared__ u64 sk[256];
  int p = blockIdx.x, t = threadIdx.x;
  u32 c = cnt[p]; if (c > KSEL) c = KSEL;
  sk[t] = ((u32)t < c) ? keys[p*KSEL + t] : 0ull;
  __syncthreads();
  // bitonic sort, descending
  for (int k = 2; k <= 256; k <<= 1){
    for (int j = k >> 1; j > 0; j >>= 1){
      int ixj = t ^ j;
      if (ixj > t){
        u64 a = sk[t], b = sk[ixj];
        bool descBlock = ((t & k) == 0);
        if (descBlock ? (a < b) : (a > b)){ sk[t] = b; sk[ixj] = a; }
      }
      __syncthreads();
    }
  }
  int sel = p % 3;
  u64 key = sk[t];
  u32 flat = (~(u32)key) & 0x1FFFFFu;
  u32 vk = (u32)(key >> 21);
  u32 m  = (sel == 1) ? vk : ~vk;
  float v = __uint_as_float(m & 0x7FFFFFFFu);
  if (key == 0ull){ v = 0.0f; flat = 0u; }
  vals[p*KSEL + t] = v;
  pidx[p*KSEL + t] = flat;
}

__global__ __launch_bounds__(256) void finalize_kernel(const float* __restrict__ vals, const u32* __restrict__ pidx,
                                                       const u32* __restrict__ gmax, float* __restrict__ out){
  __shared__ float red[256];
  int t = threadIdx.x;
  const float invS = 1.0f / (float)SDIM;
  float total = 0.0f;
  for (int i = 0; i < 16; ++i){
    int px = i*3, py = (16 + i)*3;
    float c0, c1;
    { // pd0: births = minima(asc), deaths = maxima(desc), coords from minima idx
      float bx = vals[(px+0)*KSEL + t];
      float dx = fmaxf(vals[(px+1)*KSEL + t], bx);
      u32 ia = pidx[(px+0)*KSEL + t];
      float fya = (float)(ia / SDIM) * invS, fxa = (float)(ia % SDIM) * invS;
      float by = vals[(py+0)*KSEL + t];
      float dy = fmaxf(vals[(py+1)*KSEL + t], by);
      u32 ib = pidx[(py+0)*KSEL + t];
      float fyb = (float)(ib / SDIM) * invS, fxb = (float)(ib % SDIM) * invS;
      float d1 = bx - by, d2 = dx - dy, d3 = fya - fyb, d4 = fxa - fxb;
      float cm = d1*d1 + d2*d2 + 0.5f*(d3*d3 + d4*d4);
      float e0 = (dx - bx)*0.5f, e1 = (dy - by)*0.5f;
      c0 = fminf(cm, e0*e0 + e1*e1);
    }
    { // pd1: births = maxima(asc), deaths = global max
      float bx = vals[(px+2)*KSEL + t];
      float dx = fmaxf(__uint_as_float(gmax[i] & 0x7FFFFFFFu), bx);
      u32 ia = pidx[(px+2)*KSEL + t];
      float fya = (float)(ia / SDIM) * invS, fxa = (float)(ia % SDIM) * invS;
      float by = vals[(py+2)*KSEL + t];
      float dy = fmaxf(__uint_as_float(gmax[16 + i] & 0x7FFFFFFFu), by);
      u32 ib = pidx[(py+2)*KSEL + t];
      float fyb = (float)(ib / SDIM) * invS, fxb = (float)(ib % SDIM) * invS;
      float d1 = bx - by, d2 = dx - dy, d3 = fya - fyb, d4 = fxa - fxb;
      float cm = d1*d1 + d2*d2 + 0.5f*(d3*d3 + d4*d4);
      float e0 = (dx - bx)*0.5f, e1 = (dy - by)*0.5f;
      c1 = fminf(cm, e0*e0 + e1*e1);
    }
    red[t] = c0; __syncthreads();
    for (int s2 = 128; s2 > 0; s2 >>= 1){ if (t < s2) red[t] += red[t+s2]; __syncthreads(); }
    float s0 = red[0]; __syncthreads();
    red[t] = c1; __syncthreads();
    for (int s2 = 128; s2 > 0; s2 >>= 1){ if (t < s2) red[t] += red[t+s2]; __syncthreads(); }
    if (t == 0) total += sqrtf(s0 + 1e-12f) + sqrtf(red[0] + 1e-12f);
    __syncthreads();
  }
  if (t == 0) out[0] = total * (1.0f/16.0f);
}

extern "C" void kernel_launch(void* const* d_in, const int* in_sizes, int n_in,
                              void* d_out, int out_size, void* d_ws, size_t ws_size,
                              hipStream_t stream){
  const float* in = (const float*)d_in[0];
  const float* tg = (const float*)d_in[1];
  char* ws = (char*)d_ws;
  u32* hist   = (u32*)(ws + O_HIST);
  u64* prefix = (u64*)(ws + O_PREFIX);
  u32* krem   = (u32*)(ws + O_KREM);
  u32* cnt    = (u32*)(ws + O_CNT);
  u32* gmax   = (u32*)(ws + O_GMAX);
  u64* keys   = (u64*)(ws + O_KEYS);
  float* vals = (float*)(ws + O_VALS);
  u32* pidx   = (u32*)(ws + O_PIDX);
  float* out  = (float*)d_out;

  init_kernel<<<PAIRS, 256, 0, stream>>>(hist, prefix, krem, cnt, gmax);
  gmax_kernel<<<dim3(IMG_ELEMS/256, IMGS), 256, 0, stream>>>(in, tg, gmax);
  for (int pass = 0; pass < 7; ++pass){
    int shift = 48 - 8*pass;
    hist_kernel<<<dim3(TILES, PAIRS), 256, 0, stream>>>(in, tg, hist, prefix, shift);
    scan_kernel<<<PAIRS, 256, 0, stream>>>(hist, prefix, krem, shift);
  }
  collect_kernel<<<dim3(TILES, PAIRS), 256, 0, stream>>>(in, tg, prefix, cnt, keys);
  sort_kernel<<<PAIRS, 256, 0, stream>>>(keys, cnt, vals, pidx);
  finalize_kernel<<<1, 256, 0, stream>>>(vals, pidx, gmax, out);
}